// UniGCNIIConv_88708254532193
// MI455X (gfx1250) — compile-verified
//
#include <hip/hip_runtime.h>
#include <hip/hip_bf16.h>

#define NV   50000
#define ME   10000
#define NNZC 1600000
#define DF   128

typedef __attribute__((ext_vector_type(2))) float v2f;
typedef __attribute__((ext_vector_type(8))) float v8f;

// ---- workspace layout (floats) -------------------------------------------
// [OFF_XE, +ME*DF)  Xe sum accumulator (becomes mean in place)
// [OFF_CE, +ME)     per-edge counts
// [OFF_XV, +NV*DF)  Xv sum accumulator (becomes Xi in place)
// [OFF_CV, +NV)     per-vertex counts
static constexpr size_t OFF_XE = 0;
static constexpr size_t OFF_CE = (size_t)ME * DF;
static constexpr size_t OFF_XV = OFF_CE + ME;
static constexpr size_t OFF_CV = OFF_XV + (size_t)NV * DF;
static constexpr size_t WS_FLOATS = OFF_CV + NV;   // 7,740,000 floats (~31 MB)

// ---- kernels -------------------------------------------------------------

__global__ __launch_bounds__(256) void zero_ws_kernel(float4* __restrict__ ws, int n4) {
    int t = blockIdx.x * blockDim.x + threadIdx.x;
    if (t < n4) ws[t] = make_float4(0.f, 0.f, 0.f, 0.f);
}

__global__ __launch_bounds__(256) void count_kernel(const int* __restrict__ vertex,
                                                    const int* __restrict__ edges,
                                                    float* __restrict__ cnt_e,
                                                    float* __restrict__ cnt_v, int nnz) {
    int i = blockIdx.x * blockDim.x + threadIdx.x;
    if (i < nnz) {
        atomicAdd(&cnt_e[edges[i]], 1.0f);
        atomicAdd(&cnt_v[vertex[i]], 1.0f);
    }
}

// gather X[vertex[i]] row, scatter-add into Xe[edges[i]]. One wave per nnz entry.
__global__ __launch_bounds__(256) void scatter_edge_kernel(const float* __restrict__ X,
                                                           const int* __restrict__ vertex,
                                                           const int* __restrict__ edges,
                                                           float* __restrict__ Xe, int nnz) {
    int t  = blockIdx.x * blockDim.x + threadIdx.x;
    int i  = t >> 5;                 // incidence index (shared by whole wave)
    int d4 = (t & 31) << 2;          // float4 column offset, 0..124
    if (i < nnz) {
        int v = vertex[i], e = edges[i];
        float4 x = *(const float4*)(X + (size_t)v * DF + d4);
        float* dst = Xe + (size_t)e * DF + d4;
        atomicAdd(dst + 0, x.x);
        atomicAdd(dst + 1, x.y);
        atomicAdd(dst + 2, x.z);
        atomicAdd(dst + 3, x.w);
    }
}

__global__ __launch_bounds__(256) void mean_edge_kernel(float* __restrict__ Xe,
                                                        const float* __restrict__ cnt_e) {
    int t = blockIdx.x * blockDim.x + threadIdx.x;   // ME * 32 threads
    if (t < ME * (DF / 4)) {
        int e = t >> 5;
        float inv = 1.0f / fmaxf(cnt_e[e], 1.0f);
        float4* p = (float4*)Xe + t;
        float4 x = *p;
        x.x *= inv; x.y *= inv; x.z *= inv; x.w *= inv;
        *p = x;
    }
}

// gather Xe[edges[i]] row, scatter-add into Xv[vertex[i]].
__global__ __launch_bounds__(256) void scatter_vertex_kernel(const float* __restrict__ Xe,
                                                             const int* __restrict__ vertex,
                                                             const int* __restrict__ edges,
                                                             float* __restrict__ Xv, int nnz) {
    int t  = blockIdx.x * blockDim.x + threadIdx.x;
    int i  = t >> 5;
    int d4 = (t & 31) << 2;
    if (i < nnz) {
        int v = vertex[i], e = edges[i];
        float4 x = *(const float4*)(Xe + (size_t)e * DF + d4);
        float* dst = Xv + (size_t)v * DF + d4;
        atomicAdd(dst + 0, x.x);
        atomicAdd(dst + 1, x.y);
        atomicAdd(dst + 2, x.z);
        atomicAdd(dst + 3, x.w);
    }
}

// Xi = (1-alpha) * Xv/cnt + alpha * X0, written in place over Xv.
__global__ __launch_bounds__(256) void xi_kernel(float* __restrict__ Xv,
                                                 const float* __restrict__ cnt_v,
                                                 const float* __restrict__ X0,
                                                 const float* __restrict__ alpha_p) {
    int t = blockIdx.x * blockDim.x + threadIdx.x;   // NV * 32 threads
    if (t < NV * (DF / 4)) {
        int v = t >> 5;
        float a   = *alpha_p;
        float oma = 1.0f - a;
        float inv = 1.0f / fmaxf(cnt_v[v], 1.0f);
        float4 s  = ((float4*)Xv)[t];
        float4 x0 = ((const float4*)X0)[t];
        s.x = oma * (s.x * inv) + a * x0.x;
        s.y = oma * (s.y * inv) + a * x0.y;
        s.z = oma * (s.z * inv) + a * x0.z;
        s.w = oma * (s.w * inv) + a * x0.w;
        ((float4*)Xv)[t] = s;
    }
}

// out = (1-beta)*Xi + beta*(Xi @ W^T) using V_WMMA_F32_16X16X4_F32.
// Block = 256 threads = 8 waves; block b covers rows [16b,16b+16); wave w covers
// output columns [16w,16w+16). K loop over DF in steps of 4.
__global__ __launch_bounds__(256) void gemm_mix_kernel(const float* __restrict__ Xi,
                                                       const float* __restrict__ W,
                                                       const float* __restrict__ beta_p,
                                                       float* __restrict__ out) {
    const float beta = *beta_p;
    const int lane = threadIdx.x & 31;
    const int wave = threadIdx.x >> 5;       // 0..7 -> column tile
    const int row0 = blockIdx.x * 16;
    const int col0 = wave * 16;

    // f32 WMMA A layout (16x4 MxK): lanes 0-15 hold K=0,1; lanes 16-31 hold K=2,3.
    // B (4x16 KxN) mirrors it column-major: same striping with N = lane&15.
    const int r    = lane & 15;
    const int koff = (lane >> 4) << 1;       // 0 or 2

    v8f acc = {};
#pragma unroll
    for (int k0 = 0; k0 < DF; k0 += 4) {
        const float* ap = Xi + (size_t)(row0 + r) * DF + k0 + koff;
        const float* bp = W  + (size_t)(col0 + r) * DF + k0 + koff;  // B[k][n] = W[n][k]
        v2f a, b;
        a.x = ap[0]; a.y = ap[1];
        b.x = bp[0]; b.y = bp[1];
        // 8 args: (neg_a, A, neg_b, B, c_mod, C, reuse_a, reuse_b)
        acc = __builtin_amdgcn_wmma_f32_16x16x4_f32(
            /*neg_a=*/false, a, /*neg_b=*/false, b,
            /*c_mod=*/(short)0, acc, /*reuse_a=*/false, /*reuse_b=*/false);
    }

    // C/D layout: VGPR v, lanes 0-15 -> M=v, N=lane; lanes 16-31 -> M=v+8, N=lane-16.
    const int n   = lane & 15;
    const int mhi = (lane >> 4) * 8;
    const float omb = 1.0f - beta;
#pragma unroll
    for (int v = 0; v < 8; ++v) {
        size_t o = (size_t)(row0 + mhi + v) * DF + col0 + n;
        out[o] = omb * Xi[o] + beta * acc[v];
    }
}

// ---- launcher ------------------------------------------------------------

extern "C" void kernel_launch(void* const* d_in, const int* in_sizes, int n_in,
                              void* d_out, int out_size, void* d_ws, size_t ws_size,
                              hipStream_t stream) {
    const float* X      = (const float*)d_in[0];
    const float* X0     = (const float*)d_in[1];
    const float* W      = (const float*)d_in[2];
    const float* alpha  = (const float*)d_in[3];
    const float* beta   = (const float*)d_in[4];
    const int*   vertex = (const int*)d_in[5];
    const int*   edges  = (const int*)d_in[6];
    float* out = (float*)d_out;

    float* ws    = (float*)d_ws;
    float* Xe    = ws + OFF_XE;
    float* cnt_e = ws + OFF_CE;
    float* Xv    = ws + OFF_XV;     // becomes Xi in place
    float* cnt_v = ws + OFF_CV;

    const int nnz = NNZC;
    const int B = 256;

    // 0) zero accumulators + counts
    {
        int n4 = (int)(WS_FLOATS / 4);
        zero_ws_kernel<<<(n4 + B - 1) / B, B, 0, stream>>>((float4*)d_ws, n4);
    }
    // 1) segment counts for both scatters
    count_kernel<<<(nnz + B - 1) / B, B, 0, stream>>>(vertex, edges, cnt_e, cnt_v, nnz);
    // 2) Xe += X[vertex] scattered by edge
    {
        long long t = (long long)nnz * 32;
        scatter_edge_kernel<<<(unsigned)((t + B - 1) / B), B, 0, stream>>>(X, vertex, edges, Xe, nnz);
    }
    // 3) Xe /= count
    mean_edge_kernel<<<(ME * (DF / 4) + B - 1) / B, B, 0, stream>>>(Xe, cnt_e);
    // 4) Xv += Xe[edges] scattered by vertex
    {
        long long t = (long long)nnz * 32;
        scatter_vertex_kernel<<<(unsigned)((t + B - 1) / B), B, 0, stream>>>(Xe, vertex, edges, Xv, nnz);
    }
    // 5) Xi = (1-alpha) * Xv/cnt + alpha * X0   (in place in Xv)
    xi_kernel<<<(NV * (DF / 4) + B - 1) / B, B, 0, stream>>>(Xv, cnt_v, X0, alpha);
    // 6) out = (1-beta)*Xi + beta * Xi @ W^T    (WMMA f32 16x16x4)
    gemm_mix_kernel<<<NV / 16, 256, 0, stream>>>(Xv, W, beta, out);
}